// Regional_Interaction_Module_5961414607233
// MI455X (gfx1250) — compile-verified
//
#include <hip/hip_runtime.h>
#include <hip/hip_bf16.h>

typedef __attribute__((ext_vector_type(2))) float v2f;
typedef __attribute__((ext_vector_type(8))) float v8f;

#define HW_    (512 * 512)          // pixels per plane (2^18)
#define NBATCH 32
// 16384 tiles of 16 pixels per batch plane; 1024 waves per batch, 16 tiles/wave.
#define WAVES_PER_BATCH 1024
#define TILES_PER_WAVE  16

__device__ __forceinline__ float fast_sigmoid(float x) {
    // v_exp_f32 + v_rcp_f32 (1-ulp) instead of the IEEE div_scale/div_fmas chain
    return __builtin_amdgcn_rcpf(1.0f + __expf(-x));
}

__global__ __launch_bounds__(256)
void rim_kernel(const float* __restrict__ img,  const float* __restrict__ mask,
                const float* __restrict__ qi_w, const float* __restrict__ qi_b,
                const float* __restrict__ ki_w, const float* __restrict__ ki_b,
                const float* __restrict__ vi_w, const float* __restrict__ vi_b,
                const float* __restrict__ qr_w, const float* __restrict__ qr_b,
                const float* __restrict__ kr_w, const float* __restrict__ kr_b,
                const float* __restrict__ vr_w, const float* __restrict__ vr_b,
                const float* __restrict__ fu_w, const float* __restrict__ fu_b,
                float* __restrict__ out)
{
    // ---- Packed stage-1 weights W1 (16x4, rows 8..15 zero) + biases in LDS ----
    // Input channels: 0=imgR 1=imgG 2=imgB 3=mask
    // Output rows:    0=q_i 1=k_i 2..4=v_i 5=q_r 6=k_r 7=v_r
    __shared__ float sW1[64];
    __shared__ float sB1[8];
    const int tid = threadIdx.x;
    if (tid < 64) sW1[tid] = 0.0f;
    if (tid >= 64 && tid < 72) sB1[tid - 64] = 0.0f;
    __syncthreads();
    if (tid == 0) {
        #pragma unroll
        for (int j = 0; j < 3; ++j) {
            sW1[0 * 4 + j] = qi_w[j];
            sW1[1 * 4 + j] = ki_w[j];
            sW1[2 * 4 + j] = vi_w[0 * 3 + j];
            sW1[3 * 4 + j] = vi_w[1 * 3 + j];
            sW1[4 * 4 + j] = vi_w[2 * 3 + j];
        }
        sW1[5 * 4 + 3] = qr_w[0];
        sW1[6 * 4 + 3] = kr_w[0];
        sW1[7 * 4 + 3] = vr_w[0];
        sB1[0] = qi_b[0]; sB1[1] = ki_b[0];
        sB1[2] = vi_b[0]; sB1[3] = vi_b[1]; sB1[4] = vi_b[2];
        sB1[5] = qr_b[0]; sB1[6] = kr_b[0]; sB1[7] = vr_b[0];
    }
    __syncthreads();

    // ---- Fusion conv weights: uniform addresses -> scalar loads ----
    float fw[18];
    #pragma unroll
    for (int i = 0; i < 18; ++i) fw[i] = fu_w[i];
    const float fb0 = fu_b[0], fb1 = fu_b[1], fb2 = fu_b[2];

    const int lane = tid & 31;
    const int n    = lane & 15;   // pixel-in-tile
    const int hi   = lane >> 4;   // 0 => supplies K=0,1 ; 1 => supplies K=2,3

    // A matrix (16x4 f32): lane m holds K=0,1; lane m+16 holds K=2,3
    v2f A;
    A.x = sW1[n * 4 + 2 * hi + 0];
    A.y = sW1[n * 4 + 2 * hi + 1];

    // C matrix: biases broadcast along N (rows 8..15 zero)
    v8f C;
    #pragma unroll
    for (int r = 0; r < 8; ++r) C[r] = hi ? 0.0f : sB1[r];

    // ---- Work partition: wave -> one contiguous 256-pixel span of one batch ----
    const int w     = (blockIdx.x * blockDim.x + tid) >> 5;   // 0..32767
    const int b     = w >> 10;                                // batch index
    const int chunk = w & (WAVES_PER_BATCH - 1);
    const int s0    = chunk * (TILES_PER_WAVE * 16);          // start pixel in plane

    const float* pr = img + (size_t)(3 * b) * HW_ + s0;       // R plane
    // Loop-invariant per-lane load bases (branchless select once):
    const float* q0 = (hi ? (pr + 2 * HW_) : pr) + n;                              // K=2(B)|K=0(R)
    const float* q1 = (hi ? (mask + (size_t)b * HW_ + s0) : (pr + HW_)) + n;       // K=3(m)|K=1(G)
    float*       po = out + (size_t)(3 * b) * HW_ + s0 + n;

    #pragma unroll 2
    for (int it = 0; it < TILES_PER_WAVE; ++it, q0 += 16, q1 += 16, po += 16) {
        v2f Bm;
        Bm.x = q0[0];
        Bm.y = q1[0];

        // stage1 = W1 * x + b1  (EXEC all-1s; the store-`if` below closes each iter)
        v8f D = __builtin_amdgcn_wmma_f32_16x16x4_f32(
            /*neg_a=*/false, A, /*neg_b=*/false, Bm,
            /*c_mod=*/(short)0, C, /*reuse_a=*/false, /*reuse_b=*/false);

        const float qi = D[0], ki = D[1];
        const float vi0 = D[2], vi1 = D[3], vi2 = D[4];
        const float qr = D[5], kr = D[6], vr = D[7];

        const float sii = fast_sigmoid(qi * ki);
        const float sir = fast_sigmoid(qi * kr);
        const float srr = fast_sigmoid(qr * kr);
        const float sri = fast_sigmoid(qr * ki);

        const float fi0 = sii * vi0 + sir * vr;
        const float fi1 = sii * vi1 + sir * vr;
        const float fi2 = sii * vi2 + sir * vr;
        const float fr0 = srr * vr + sri * vi0;
        const float fr1 = srr * vr + sri * vi1;
        const float fr2 = srr * vr + sri * vi2;

        const float o0 = fw[0]*fi0 + fw[1]*fi1 + fw[2]*fi2 + fw[3]*fr0 + fw[4]*fr1 + fw[5]*fr2 + fb0;
        const float o1 = fw[6]*fi0 + fw[7]*fi1 + fw[8]*fi2 + fw[9]*fr0 + fw[10]*fr1 + fw[11]*fr2 + fb1;
        const float o2 = fw[12]*fi0 + fw[13]*fi1 + fw[14]*fi2 + fw[15]*fr0 + fw[16]*fr1 + fw[17]*fr2 + fb2;

        if (!hi) {                     // lanes 0-15 each own one pixel; coalesced 64B runs
            po[0]       = o0;
            po[HW_]     = o1;
            po[2 * HW_] = o2;
        }
    }
}

extern "C" void kernel_launch(void* const* d_in, const int* in_sizes, int n_in,
                              void* d_out, int out_size, void* d_ws, size_t ws_size,
                              hipStream_t stream) {
    (void)in_sizes; (void)n_in; (void)out_size; (void)d_ws; (void)ws_size;
    const float* img  = (const float*)d_in[0];
    const float* mask = (const float*)d_in[1];
    const float* qi_w = (const float*)d_in[2];
    const float* qi_b = (const float*)d_in[3];
    const float* ki_w = (const float*)d_in[4];
    const float* ki_b = (const float*)d_in[5];
    const float* vi_w = (const float*)d_in[6];
    const float* vi_b = (const float*)d_in[7];
    const float* qr_w = (const float*)d_in[8];
    const float* qr_b = (const float*)d_in[9];
    const float* kr_w = (const float*)d_in[10];
    const float* kr_b = (const float*)d_in[11];
    const float* vr_w = (const float*)d_in[12];
    const float* vr_b = (const float*)d_in[13];
    const float* fu_w = (const float*)d_in[14];
    const float* fu_b = (const float*)d_in[15];
    float* out = (float*)d_out;

    // Exactly 32768 waves = 32 batches x 1024 waves, each covering 16 tiles.
    dim3 grid(4096), block(256);
    rim_kernel<<<grid, block, 0, stream>>>(img, mask, qi_w, qi_b, ki_w, ki_b,
                                           vi_w, vi_b, qr_w, qr_b, kr_w, kr_b,
                                           vr_w, vr_b, fu_w, fu_b, out);
}